// HypergraphBlock_20220706030436
// MI455X (gfx1250) — compile-verified
//
#include <hip/hip_runtime.h>
#include <hip/hip_bf16.h>

// ---------------------------------------------------------------------------
// HypergraphConv + transformer-ish block for MI455X (gfx1250, wave32).
//
// Shapes: N=50000 nodes, NUM_HE=50000 hyperedges, E_INC=800000 incidences,
//         D=128, FF=256.  All tile dims divide evenly (50000/16=3125).
//
// Scatter phases are the bandwidth/atomic bound (~1.6GB -> ~70us at
// 23.3TB/s; atomics resolve in the 192MB L2). GEMMs are a small ~8.2 GFLOP
// fp32 job -> exact-precision V_WMMA_F32_16X16X4_F32, with all strides as
// compile-time constants so inner-loop addresses are immediate offsets
// (no 64-bit multiplies per load).
// ---------------------------------------------------------------------------

#define NN      50000
#define NUM_HE  50000
#define DD      128
#define FF      256

typedef float __attribute__((ext_vector_type(2))) v2f;
typedef float __attribute__((ext_vector_type(8))) v8f;

// ---------------------------------------------------------------------------
// GEMM: C[M,NCOLS] = act(A[M,K] @ B[K,NCOLS] + bias)
// Block = 256 threads = 8 waves; block covers 16 rows x 128 cols
// (one 16x16 tile per wave); grid.y covers NCOLS/128.
// f32 WMMA fragment layout (wave32):
//   A 16x4 : lane L -> M=L&15, K = k + 2*(L>>4) + {0,1}  (v2f, b64 load)
//   B 4x16 : lane L -> N=L&15, K = k + 2*(L>>4) + {0,1}  (2x b32, imm offs)
//   C 16x16: vgpr r -> M = r + 8*(L>>4), N = L&15        (v8f)
// K*NCOLS*4 <= 256*128*4 = 128KB fits the signed 24-bit global IOFFSET.
// ---------------------------------------------------------------------------
template<int K, int NCOLS, int RELU, int HAS_BIAS>
__global__ __launch_bounds__(256)
void gemm_wmma_f32(const float* __restrict__ A, const float* __restrict__ B,
                   const float* __restrict__ bias, float* __restrict__ C)
{
    const int wave = threadIdx.x >> 5;
    const int lane = threadIdx.x & 31;
    const int l15  = lane & 15;
    const int half = lane >> 4;          // 0: K+{0,1}, 1: K+{2,3}
    const int koff = half * 2;

    const int row0 = blockIdx.x * 16;
    const int col0 = blockIdx.y * 128 + wave * 16;

    // A pair is 8B-aligned: row base is K*4B aligned, (k+koff) is even.
    const float* __restrict__ arow = A + (size_t)(row0 + l15) * K + koff;
    const float* __restrict__ bcol = B + (size_t)koff * NCOLS + col0 + l15;

    v8f acc = {};
#pragma unroll
    for (int k = 0; k < K; k += 4) {
        v2f a = *(const v2f*)(arow + k);               // global_load_b64, imm off
        v2f b;
        b.x = bcol[(size_t)k * NCOLS];                 // global_load_b32, imm off
        b.y = bcol[(size_t)k * NCOLS + NCOLS];         // global_load_b32, imm off
        acc = __builtin_amdgcn_wmma_f32_16x16x4_f32(
                  /*neg_a=*/false, a, /*neg_b=*/false, b,
                  /*c_mod=*/(short)0, acc, /*reuse_a=*/false, /*reuse_b=*/false);
    }

    const int col = col0 + l15;
    float bv = 0.0f;
    if (HAS_BIAS) bv = bias[col];

    float* __restrict__ crow = C + (size_t)(row0 + half * 8) * NCOLS + col;
#pragma unroll
    for (int r = 0; r < 8; ++r) {
        float v = acc[r] + bv;
        if (RELU) v = fmaxf(v, 0.0f);
        crow[(size_t)r * NCOLS] = v;                   // imm-offset stores
    }
}

// ---------------------------------------------------------------------------
// Degrees: Bdeg[he] += 1 ; Ddeg[node] += edge_weight[he]
// ---------------------------------------------------------------------------
__global__ __launch_bounds__(256)
void degrees_kernel(const int* __restrict__ node_idx, const int* __restrict__ he_idx,
                    const float* __restrict__ ew,
                    float* __restrict__ Bdeg, float* __restrict__ Ddeg, int ninc)
{
    int e = blockIdx.x * 256 + threadIdx.x;
    if (e >= ninc) return;
    int h = he_idx[e];
    int n = node_idx[e];
    atomicAdd(&Bdeg[h], 1.0f);
    atomicAdd(&Ddeg[n], ew[h]);
}

// ---------------------------------------------------------------------------
// Scatter node->hyperedge: macc[he] += xw[node]. One wave per incidence
// entry; each lane owns a float4 of the 128-wide row (32*4 = 128).
// ---------------------------------------------------------------------------
__global__ __launch_bounds__(256)
void scatter_n2e_kernel(const float* __restrict__ xw,
                        const int* __restrict__ node_idx, const int* __restrict__ he_idx,
                        float* __restrict__ macc, int ninc)
{
    int e = blockIdx.x * 8 + (threadIdx.x >> 5);
    if (e >= ninc) return;
    int lane = threadIdx.x & 31;
    int n = node_idx[e];
    int h = he_idx[e];
    float4 v = ((const float4*)(xw + (size_t)n * DD))[lane];
    float* d = macc + (size_t)h * DD + lane * 4;
    atomicAdd(d + 0, v.x);
    atomicAdd(d + 1, v.y);
    atomicAdd(d + 2, v.z);
    atomicAdd(d + 3, v.w);
}

// ---------------------------------------------------------------------------
// Scatter hyperedge->node with fused (Binv * w) scaling:
//   cacc[node] += macc[he] * (w[he]/Bdeg[he])
// ---------------------------------------------------------------------------
__global__ __launch_bounds__(256)
void scatter_e2n_kernel(const float* __restrict__ macc,
                        const float* __restrict__ Bdeg, const float* __restrict__ ew,
                        const int* __restrict__ node_idx, const int* __restrict__ he_idx,
                        float* __restrict__ cacc, int ninc)
{
    int e = blockIdx.x * 8 + (threadIdx.x >> 5);
    if (e >= ninc) return;
    int lane = threadIdx.x & 31;
    int n = node_idx[e];
    int h = he_idx[e];
    float bd = Bdeg[h];
    float sc = bd > 0.0f ? ew[h] / bd : 0.0f;
    float4 v = ((const float4*)(macc + (size_t)h * DD))[lane];
    float* d = cacc + (size_t)n * DD + lane * 4;
    atomicAdd(d + 0, v.x * sc);
    atomicAdd(d + 1, v.y * sc);
    atomicAdd(d + 2, v.z * sc);
    atomicAdd(d + 3, v.w * sc);
}

// ---------------------------------------------------------------------------
// Wave32 reduction helper.
// ---------------------------------------------------------------------------
__device__ __forceinline__ float wave_sum32(float s)
{
#pragma unroll
    for (int off = 16; off > 0; off >>= 1)
        s += __shfl_xor(s, off, 32);
    return s;
}

// ---------------------------------------------------------------------------
// Fused: conv = cacc*Dinv + bc ; h = relu(LN(conv)*g1+be1) + x
// One wave (32 lanes) per 128-wide row; 8 rows per 256-thread block.
// ---------------------------------------------------------------------------
__global__ __launch_bounds__(256)
void conv_ln_relu_res_kernel(const float* __restrict__ cacc,
                             const float* __restrict__ Ddeg,
                             const float* __restrict__ bc,
                             const float* __restrict__ g,  const float* __restrict__ be,
                             const float* __restrict__ xres,
                             float* __restrict__ hout, int nrows)
{
    int row = blockIdx.x * 8 + (threadIdx.x >> 5);
    if (row >= nrows) return;
    int lane = threadIdx.x & 31;

    float dd   = Ddeg[row];
    float dinv = dd > 0.0f ? 1.0f / dd : 0.0f;

    float4 a  = ((const float4*)(cacc + (size_t)row * DD))[lane];
    float4 bb = ((const float4*)bc)[lane];
    float4 v;
    v.x = a.x * dinv + bb.x;
    v.y = a.y * dinv + bb.y;
    v.z = a.z * dinv + bb.z;
    v.w = a.w * dinv + bb.w;

    float mu = wave_sum32(v.x + v.y + v.z + v.w) * (1.0f / 128.0f);
    float dx = v.x - mu, dy = v.y - mu, dz = v.z - mu, dw = v.w - mu;
    float var  = wave_sum32(dx * dx + dy * dy + dz * dz + dw * dw) * (1.0f / 128.0f);
    float rstd = rsqrtf(var + 1e-5f);

    float4 gg = ((const float4*)g)[lane];
    float4 ee = ((const float4*)be)[lane];
    float4 rr = ((const float4*)(xres + (size_t)row * DD))[lane];
    float4 o;
    o.x = fmaxf(dx * rstd * gg.x + ee.x, 0.0f) + rr.x;
    o.y = fmaxf(dy * rstd * gg.y + ee.y, 0.0f) + rr.y;
    o.z = fmaxf(dz * rstd * gg.z + ee.z, 0.0f) + rr.z;
    o.w = fmaxf(dw * rstd * gg.w + ee.w, 0.0f) + rr.w;
    ((float4*)(hout + (size_t)row * DD))[lane] = o;
}

// ---------------------------------------------------------------------------
// Fused final: out = LN(f)*g2+be2 + h
// ---------------------------------------------------------------------------
__global__ __launch_bounds__(256)
void ln_res_kernel(const float* __restrict__ f,
                   const float* __restrict__ g,  const float* __restrict__ be,
                   const float* __restrict__ hres,
                   float* __restrict__ out, int nrows)
{
    int row = blockIdx.x * 8 + (threadIdx.x >> 5);
    if (row >= nrows) return;
    int lane = threadIdx.x & 31;

    float4 v = ((const float4*)(f + (size_t)row * DD))[lane];
    float mu = wave_sum32(v.x + v.y + v.z + v.w) * (1.0f / 128.0f);
    float dx = v.x - mu, dy = v.y - mu, dz = v.z - mu, dw = v.w - mu;
    float var  = wave_sum32(dx * dx + dy * dy + dz * dz + dw * dw) * (1.0f / 128.0f);
    float rstd = rsqrtf(var + 1e-5f);

    float4 gg = ((const float4*)g)[lane];
    float4 ee = ((const float4*)be)[lane];
    float4 rr = ((const float4*)(hres + (size_t)row * DD))[lane];
    float4 o;
    o.x = dx * rstd * gg.x + ee.x + rr.x;
    o.y = dy * rstd * gg.y + ee.y + rr.y;
    o.z = dz * rstd * gg.z + ee.z + rr.z;
    o.w = dw * rstd * gg.w + ee.w + rr.w;
    ((float4*)(out + (size_t)row * DD))[lane] = o;
}

// ---------------------------------------------------------------------------
// Host launcher
// ---------------------------------------------------------------------------
extern "C" void kernel_launch(void* const* d_in, const int* in_sizes, int n_in,
                              void* d_out, int out_size, void* d_ws, size_t ws_size,
                              hipStream_t stream)
{
    const float* x    = (const float*)d_in[0];
    const int*   eidx = (const int*)  d_in[1];   // [2, E_INC]: row0=node, row1=he
    const float* ew   = (const float*)d_in[2];
    const float* Wc   = (const float*)d_in[3];
    const float* bc   = (const float*)d_in[4];
    const float* g1   = (const float*)d_in[5];
    const float* be1  = (const float*)d_in[6];
    const float* W1   = (const float*)d_in[7];
    const float* bf1  = (const float*)d_in[8];
    const float* W2   = (const float*)d_in[9];
    const float* bf2  = (const float*)d_in[10];
    const float* g2   = (const float*)d_in[11];
    const float* be2  = (const float*)d_in[12];
    float*       out  = (float*)d_out;

    const int ninc = in_sizes[1] / 2;
    const int* node_idx = eidx;
    const int* he_idx   = eidx + ninc;

    // Workspace layout (fp32, all float4-aligned):
    float* ws   = (float*)d_ws;
    float* xw   = ws;  ws += (size_t)NN * DD;        // 6.4M
    float* Bdeg = ws;  ws += NUM_HE;                 // 50k   (contiguous with Ddeg)
    float* Ddeg = ws;  ws += NN;                     // 50k
    float* macc = ws;  ws += (size_t)NUM_HE * DD;    // 6.4M
    float* cacc = ws;  ws += (size_t)NN * DD;        // 6.4M
    float* hbuf = ws;  ws += (size_t)NN * DD;        // 6.4M
    float* f1   = ws;  ws += (size_t)NN * FF;        // 12.8M
    float* fbuf = ws;  ws += (size_t)NN * DD;        // 6.4M  total ~180MB

    // Zero the atomic accumulators (Bdeg+Ddeg are contiguous).
    hipMemsetAsync(Bdeg, 0, (size_t)(NUM_HE + NN) * sizeof(float), stream);
    hipMemsetAsync(macc, 0, (size_t)NUM_HE * DD * sizeof(float), stream);
    hipMemsetAsync(cacc, 0, (size_t)NN * DD * sizeof(float), stream);

    // 1. degrees
    degrees_kernel<<<(ninc + 255) / 256, 256, 0, stream>>>(
        node_idx, he_idx, ew, Bdeg, Ddeg, ninc);

    // 2. xw = x @ Wc           (M=50000, K=128, N=128)
    gemm_wmma_f32<DD, DD, 0, 0><<<dim3(NN / 16, 1), 256, 0, stream>>>(
        x, Wc, nullptr, xw);

    // 3. macc[he] += xw[node]
    scatter_n2e_kernel<<<(ninc + 7) / 8, 256, 0, stream>>>(
        xw, node_idx, he_idx, macc, ninc);

    // 4. cacc[node] += macc[he] * (w[he]/Bdeg[he])
    scatter_e2n_kernel<<<(ninc + 7) / 8, 256, 0, stream>>>(
        macc, Bdeg, ew, node_idx, he_idx, cacc, ninc);

    // 5. h = relu(LN(cacc*Dinv + bc)) + x
    conv_ln_relu_res_kernel<<<(NN + 7) / 8, 256, 0, stream>>>(
        cacc, Ddeg, bc, g1, be1, x, hbuf, NN);

    // 6. f1 = relu(h @ W1 + bf1)   (K=128, N=256)
    gemm_wmma_f32<DD, FF, 1, 1><<<dim3(NN / 16, FF / 128), 256, 0, stream>>>(
        hbuf, W1, bf1, f1);

    // 7. f = f1 @ W2 + bf2         (K=256, N=128)
    gemm_wmma_f32<FF, DD, 0, 1><<<dim3(NN / 16, 1), 256, 0, stream>>>(
        f1, W2, bf2, fbuf);

    // 8. out = LN(f) + h
    ln_res_kernel<<<(NN + 7) / 8, 256, 0, stream>>>(
        fbuf, g2, be2, hbuf, out, NN);
}